// LGCN_66219805769753
// MI455X (gfx1250) — compile-verified
//
#include <hip/hip_runtime.h>
#include <hip/hip_bf16.h>
#include <cstdint>

// ---------------------------------------------------------------------------
// LightGCN propagation on MI455X (gfx1250, wave32).
//
// Roofline: 3 layers x E*D*2 flops ~ 1.2 GFLOP vs ~5 GB of gather/scatter
// traffic -> 0.08 FLOP/byte: purely memory-bound, WMMA not applicable
// (random scatter-add has no shared-B matmul tiling). X (51 MB) is resident
// in the 192 MB L2, so the levers are:
//   - CDNA5 async-to-LDS edge streaming (ASYNCcnt double-buffer)
//   - coalesced b128 row gathers (one D=128 row per wave32)
//   - non-returning global_atomic_add_f32 scatter (STOREcnt path)
//   - minimal number of full-array passes (fused zero/accumulate/scale)
// ---------------------------------------------------------------------------

#define WPB 8          // waves per 256-thread block
#define TILE 32        // edges per wave tile (== wave32 width)

__global__ void k_zero(float* __restrict__ p, int n) {
    int i = blockIdx.x * blockDim.x + threadIdx.x;
    if (i < n) p[i] = 0.0f;
}

__global__ void k_deg(const int* __restrict__ col, float* __restrict__ deg, int E) {
    int i = blockIdx.x * blockDim.x + threadIdx.x;
    if (i < E) atomicAdd(&deg[col[i]], 1.0f);
}

__global__ void k_rsq(float* __restrict__ d, int n) {
    int i = blockIdx.x * blockDim.x + threadIdx.x;
    if (i < n) {
        float v = d[i];
        d[i] = (v > 0.0f) ? rsqrtf(v) : 0.0f;
    }
}

__global__ void k_norm(const int* __restrict__ row, const int* __restrict__ col,
                       const float* __restrict__ dis, float* __restrict__ nrm, int E) {
    int i = blockIdx.x * blockDim.x + threadIdx.x;
    if (i < E) nrm[i] = dis[row[i]] * dis[col[i]];
}

__global__ void k_copy2(const float* __restrict__ src, float* __restrict__ a,
                        float* __restrict__ b, int n) {
    int i = blockIdx.x * blockDim.x + threadIdx.x;
    if (i < n) { float v = src[i]; a[i] = v; b[i] = v; }
}

// acc += x; and zero the retired buffer so it is ready as the next layer's
// scatter destination (saves a separate full zero pass).
__global__ void k_acc_zero(float* __restrict__ acc, const float* __restrict__ x,
                           float* __restrict__ dead, int n) {
    int i = blockIdx.x * blockDim.x + threadIdx.x;
    if (i < n) { acc[i] += x[i]; dead[i] = 0.0f; }
}

// final layer: acc = (acc + x) * 1/(L+1)  (fuses the mean into the last add)
__global__ void k_acc_final(float* __restrict__ acc, const float* __restrict__ x,
                            float s, int n) {
    int i = blockIdx.x * blockDim.x + threadIdx.x;
    if (i < n) acc[i] = (acc[i] + x[i]) * s;
}

// One wave per edge tile: 32 lanes x float4 == one D=128 row per wave.
// Edge lists (row/col/norm) stream through LDS via CDNA5 async-to-LDS loads,
// double-buffered on ASYNCcnt: the next tile's 3 loads are in flight while
// the wave processes the current tile's atomic-heavy inner loop.
__global__ void __launch_bounds__(256)
k_scatter(const int* __restrict__ row, const int* __restrict__ col,
          const float* __restrict__ nrm, const float* __restrict__ xin,
          float* __restrict__ xout, int E) {
    __shared__ int   sRow[2][WPB][TILE];
    __shared__ int   sCol[2][WPB][TILE];
    __shared__ float sNrm[2][WPB][TILE];

    const int lane = threadIdx.x & 31;
    const int w    = threadIdx.x >> 5;
    const int wavesTotal = (gridDim.x * blockDim.x) >> 5;
    const int wid  = blockIdx.x * (blockDim.x >> 5) + w;
    const int nTiles = (E + TILE - 1) / TILE;
    if (wid >= nTiles) return;

    auto issue = [&](int tile, int buf) {
        int e = tile * TILE + lane;
        if (e >= E) e = E - 1;          // clamp: tail lanes load garbage, never read
        int boff = e * 4;               // byte offset into each stream
        unsigned aRow = (unsigned)(uintptr_t)&sRow[buf][w][lane];
        unsigned aCol = (unsigned)(uintptr_t)&sCol[buf][w][lane];
        unsigned aNrm = (unsigned)(uintptr_t)&sNrm[buf][w][lane];
        asm volatile("global_load_async_to_lds_b32 %0, %1, %2"
                     :: "v"(aRow), "v"(boff), "s"(row) : "memory");
        asm volatile("global_load_async_to_lds_b32 %0, %1, %2"
                     :: "v"(aCol), "v"(boff), "s"(col) : "memory");
        asm volatile("global_load_async_to_lds_b32 %0, %1, %2"
                     :: "v"(aNrm), "v"(boff), "s"(nrm) : "memory");
    };

    const int laneOff = lane << 2;      // this lane's 4-dim slice of the row

    int buf = 0;
    issue(wid, buf);
    for (int t = wid; t < nTiles; t += wavesTotal) {
        const int tn = t + wavesTotal;
        const bool pf = (tn < nTiles);
        if (pf) {
            issue(tn, buf ^ 1);
            // current tile's 3 loads are the oldest 3 of 6 outstanding;
            // async loads complete in order, so <=3 releases them.
            asm volatile("s_wait_asynccnt 3" ::: "memory");
        } else {
            asm volatile("s_wait_asynccnt 0" ::: "memory");
        }

        const int*   tRow = sRow[buf][w];
        const int*   tCol = sCol[buf][w];
        const float* tNrm = sNrm[buf][w];

        auto edge = [&](int e) {
            const int   r = tRow[e];    // LDS broadcast reads
            const int   c = tCol[e];
            const float f = tNrm[e];
            const float4 v = *(const float4*)(xin + ((size_t)r << 7) + laneOff);
            float* dst = xout + ((size_t)c << 7) + laneOff;
            atomicAdd(dst + 0, f * v.x);   // non-returning global_atomic_add_f32
            atomicAdd(dst + 1, f * v.y);
            atomicAdd(dst + 2, f * v.z);
            atomicAdd(dst + 3, f * v.w);
        };

        const int base = t * TILE;
        const int cnt  = min(TILE, E - base);
        if (cnt == TILE) {
            #pragma unroll 4
            for (int e = 0; e < TILE; ++e) edge(e);   // unrolled: multiple b128
        } else {                                      // gathers in flight
            for (int e = 0; e < cnt; ++e) edge(e);
        }
        buf ^= 1;
    }
}

extern "C" void kernel_launch(void* const* d_in, const int* in_sizes, int n_in,
                              void* d_out, int out_size, void* d_ws, size_t ws_size,
                              hipStream_t stream) {
    const float* emb = (const float*)d_in[0];
    const int*   ei  = (const int*)d_in[1];

    const int ND = in_sizes[0];        // N * 128
    const int E  = in_sizes[1] / 2;
    const int N  = ND / 128;
    const int* row = ei;
    const int* col = ei + E;

    float* acc = (float*)d_out;

    // carve workspace: dis[N], nrm[E], xa[ND], xb[ND]  (~110 MB total)
    uintptr_t ws = (uintptr_t)d_ws;
    auto carve = [&](size_t bytes) -> float* {
        uintptr_t p = (ws + 255) & ~(uintptr_t)255;
        ws = p + bytes;
        return (float*)p;
    };
    float* dis = carve((size_t)N * 4);
    float* nrm = carve((size_t)E * 4);
    float* xa  = carve((size_t)ND * 4);
    float* xb  = carve((size_t)ND * 4);

    const int T = 256;
    const int bN  = (N  + T - 1) / T;
    const int bE  = (E  + T - 1) / T;
    const int bND = (ND + T - 1) / T;
    const int bScat = 2048;            // 16384 waves grid-striding 50k tiles

    // degree -> deg_inv_sqrt -> per-edge norm (computed once, reused 3 layers)
    k_zero<<<bN, T, 0, stream>>>(dis, N);
    k_deg <<<bE, T, 0, stream>>>(col, dis, E);
    k_rsq <<<bN, T, 0, stream>>>(dis, N);
    k_norm<<<bE, T, 0, stream>>>(row, col, dis, nrm, E);

    // x = emb ; acc = emb ; first scatter destination zeroed
    k_copy2<<<bND, T, 0, stream>>>(emb, xa, acc, ND);
    k_zero <<<bND, T, 0, stream>>>(xb, ND);

    // 3 propagation layers; zeroing of the next destination is fused into
    // the accumulate pass, and the 1/(L+1) mean into the last one.
    for (int l = 0; l < 3; ++l) {
        k_scatter<<<bScat, T, 0, stream>>>(row, col, nrm, xa, xb, E);
        if (l < 2) {
            k_acc_zero<<<bND, T, 0, stream>>>(acc, xb, xa, ND);
        } else {
            k_acc_final<<<bND, T, 0, stream>>>(acc, xb, 0.25f, ND);
        }
        float* tmp = xa; xa = xb; xb = tmp;
    }
}